// DiagnosableGMM_36361193128735
// MI455X (gfx1250) — compile-verified
//
#include <hip/hip_runtime.h>
#include <math.h>
#include <stdint.h>

typedef float v2f __attribute__((ext_vector_type(2)));
typedef float v8f __attribute__((ext_vector_type(8)));

#define LOG2PI        1.8378770664093453f
#define PRIOR_LV0     (-2.0f)
#define INV_VAR0      7.389056098930650f   /* exp(2.0) */
#define N_ROWS        262144

// ---------------- 64-thread block reductions (setup kernel) ----------------
__device__ __forceinline__ float blk64_sum(float* sred, float v, int t) {
  sred[t] = v; __syncthreads();
  #pragma unroll
  for (int off = 32; off > 0; off >>= 1) {
    if (t < off) sred[t] = sred[t] + sred[t + off];
    __syncthreads();
  }
  float r = sred[0]; __syncthreads();
  return r;
}

__device__ __forceinline__ float blk64_max(float* sred, float v, int t) {
  sred[t] = v; __syncthreads();
  #pragma unroll
  for (int off = 32; off > 0; off >>= 1) {
    if (t < off) sred[t] = fmaxf(sred[t], sred[t + off]);
    __syncthreads();
  }
  float r = sred[0]; __syncthreads();
  return r;
}

// ---------------- setup: tiny O(K*D) precompute -----------------------------
// Produces:
//   d_out[0]    = kl_phi * N        (main kernel atomically subtracts sum lse)
//   d_out[1+t]  = q_phi[t]
//   wfrag[4096] = float2 B-fragments of W (128x64) in WMMA lane order
//   wsC[64]     = per-cluster additive constant
//   wsLp[64]    = log(softmax(pi) + 1e-9)
__global__ void gmm_setup(const float* __restrict__ u_noise,
                          const float* __restrict__ phi_logits,
                          const float* __restrict__ q_mu,
                          const float* __restrict__ q_logvar,
                          const float* __restrict__ pi_logits,
                          const float* __restrict__ prior_phi,
                          float* __restrict__ out,
                          float2* __restrict__ wfrag,
                          float* __restrict__ wsC,
                          float* __restrict__ wsLp) {
  __shared__ float sphi[64];
  __shared__ float sred[64];
  const int t = threadIdx.x;

  // Gumbel-sigmoid sample of the per-dimension gate phi
  float u   = u_noise[t];
  float gum = -logf(-logf(u + 1e-9f) + 1e-9f);
  float pl  = phi_logits[t];
  float phi = 1.0f / (1.0f + expf(-(pl + gum)));   // TEMPERATURE == 1
  sphi[t] = phi;

  // q_phi output + KL(q_phi || prior)
  float qp = 1.0f / (1.0f + expf(-pl));
  qp = fminf(fmaxf(qp, 1e-6f), 1.0f - 1e-6f);
  out[1 + t] = qp;
  float pp = fminf(fmaxf(prior_phi[t], 1e-6f), 1.0f - 1e-6f);
  float kl = qp * (logf(qp) - logf(pp))
           + (1.0f - qp) * (logf(1.0f - qp) - logf(1.0f - pp));
  float klsum = blk64_sum(sred, kl, t);
  if (t == 0) out[0] = klsum * (float)N_ROWS;

  float S1 = blk64_sum(sred, 1.0f - phi, t);       // sum_d (1 - phi_d)

  // log(softmax(pi_logits) + 1e-9)
  float l  = pi_logits[t];
  float mx = blk64_max(sred, l, t);
  float e  = expf(l - mx);
  float se = blk64_sum(sred, e, t);
  wsLp[t]  = logf(e / se + 1e-9f);

  // C[k]: per-cluster constant (cluster const + mu^2 term + background const)
  {
    float acc = 0.0f;
    for (int d = 0; d < 64; ++d) {
      float lv = fminf(fmaxf(q_logvar[t * 64 + d], -5.0f), 5.0f);
      float iv = expf(-lv);
      float a  = sphi[d] * iv;
      float mu = q_mu[t * 64 + d];
      acc += sphi[d] * (LOG2PI + lv) + mu * mu * a;
    }
    wsC[t] = -0.5f * acc - 0.5f * S1 * (LOG2PI + PRIOR_LV0);
  }

  // B fragments for v_wmma_f32_16x16x4_f32:
  //   lane l holds column c = 16*j + (l%16), K-pair r = 4*s + 2*(l/16) .. +1
  //   W[r,c]: r<64  -> Wxx[c][r]   (coef of X^2, background folded in)
  //           r>=64 -> Wx [c][r-64] (coef of X)
  for (int i = t; i < 4096; i += 64) {
    int lane = i & 31;
    int j    = (i >> 5) & 3;
    int s    = i >> 7;
    int g    = lane >> 4;
    int c    = 16 * j + (lane & 15);
    int r    = 4 * s + 2 * g;
    float v2[2];
    #pragma unroll
    for (int p = 0; p < 2; ++p) {
      int rr = r + p;
      int d  = (rr < 64) ? rr : rr - 64;
      float lv = fminf(fmaxf(q_logvar[c * 64 + d], -5.0f), 5.0f);
      float a  = sphi[d] * expf(-lv);
      v2[p] = (rr < 64) ? (-0.5f * a - 0.5f * INV_VAR0 * (1.0f - sphi[d]))
                        : (a * q_mu[c * 64 + d]);
    }
    wfrag[i] = make_float2(v2[0], v2[1]);
  }
}

// ---------------- main: fused (N x 128) @ (128 x 64) WMMA GEMM --------------
__global__ void __launch_bounds__(256)
gmm_main(const float* __restrict__ X,
         const float4* __restrict__ wfrag4,
         const float* __restrict__ wsC,
         const float* __restrict__ wsLp,
         float* __restrict__ out) {
  __shared__ float2 sW[4096];      // 32 KB: B fragments, linear per-lane order
  __shared__ float  sC[64];
  __shared__ float  sLp[64];
  __shared__ float  sPart[8];

  const int tid  = threadIdx.x;
  const int lane = tid & 31;
  const int wave = tid >> 5;
  const int g    = lane >> 4;      // half-wave group
  const int m    = lane & 15;      // row-within-tile / col-within-tile
  const int rowbase = blockIdx.x * 128 + wave * 16;

  // warm this block's X tile (gfx1250 global_prefetch_b8)
  if (tid < 128)
    __builtin_prefetch(X + (size_t)(blockIdx.x * 128 + tid) * 64, 0, 0);

  // stage weights into LDS via the gfx1250 async-copy engine (ASYNCcnt path):
  // no VGPR round trip, overlaps with the sC/sLp loads below.
  float4* sW4 = (float4*)sW;
  #pragma unroll
  for (int i = 0; i < 8; ++i) {
    uint32_t lds_dst = (uint32_t)(uintptr_t)(&sW4[tid + 256 * i]);
    uint32_t goff    = (uint32_t)((tid + 256 * i) * sizeof(float4));
    asm volatile("global_load_async_to_lds_b128 %0, %1, %2 offset:0"
                 :: "v"(lds_dst), "v"(goff), "s"(wfrag4) : "memory");
  }
  if (tid < 64) { sC[tid] = wsC[tid]; sLp[tid] = wsLp[tid]; }
#if __has_builtin(__builtin_amdgcn_s_wait_asynccnt)
  __builtin_amdgcn_s_wait_asynccnt(0);
#else
  asm volatile("s_wait_asynccnt 0x0" ::: "memory");
#endif
  __syncthreads();

  // A-fragment source: lane holds row (rowbase+m); pair t covers columns
  // {4t+2g, 4t+2g+1}. Squares feed steps 0..15, raw values steps 16..31.
  const float2* xr = (const float2*)(X + (size_t)(rowbase + m) * 64 + 2 * g);
  float2 xp[16];
  #pragma unroll
  for (int p = 0; p < 16; ++p) xp[p] = xr[2 * p];

  v8f acc[4];
  #pragma unroll
  for (int j = 0; j < 4; ++j)
    #pragma unroll
    for (int v = 0; v < 8; ++v) acc[j][v] = 0.0f;

  #pragma unroll
  for (int s = 0; s < 32; ++s) {
    v2f a;
    if (s < 16) { float2 x = xp[s];      a[0] = x.x * x.x; a[1] = x.y * x.y; }
    else        { float2 x = xp[s - 16]; a[0] = x.x;       a[1] = x.y;       }
    #pragma unroll
    for (int j = 0; j < 4; ++j) {
      float2 w = sW[(s * 4 + j) * 32 + lane];      // ds_load_2addr_b64 pairs
      v2f b; b[0] = w.x; b[1] = w.y;
      acc[j] = __builtin_amdgcn_wmma_f32_16x16x4_f32(
          false, a, false, b, (short)0, acc[j], false, false);
    }
  }

  // epilogue: add C[k], store log_p, per-row logsumexp(log_p + log pi)
  float cj[4], lpj[4];
  #pragma unroll
  for (int j = 0; j < 4; ++j) { cj[j] = sC[16 * j + m]; lpj[j] = sLp[16 * j + m]; }

  float* lp_out = out + 65;
  #pragma unroll
  for (int j = 0; j < 4; ++j)
    #pragma unroll
    for (int v = 0; v < 8; ++v) {
      float val = acc[j][v] + cj[j];
      acc[j][v] = val;
      lp_out[(size_t)(rowbase + v + 8 * g) * 64 + 16 * j + m] = val;
    }

  float part = 0.0f;
  #pragma unroll
  for (int v = 0; v < 8; ++v) {
    float z0 = acc[0][v] + lpj[0];
    float z1 = acc[1][v] + lpj[1];
    float z2 = acc[2][v] + lpj[2];
    float z3 = acc[3][v] + lpj[3];
    float mx = fmaxf(fmaxf(z0, z1), fmaxf(z2, z3));
    #pragma unroll
    for (int msk = 1; msk < 16; msk <<= 1)
      mx = fmaxf(mx, __shfl_xor(mx, msk, 16));
    float sm = expf(z0 - mx) + expf(z1 - mx) + expf(z2 - mx) + expf(z3 - mx);
    #pragma unroll
    for (int msk = 1; msk < 16; msk <<= 1)
      sm += __shfl_xor(sm, msk, 16);
    if (m == 0) part += mx + logf(sm);
  }
  part += __shfl_xor(part, 16, 32);      // combine the two half-wave leaders
  if (lane == 0) sPart[wave] = part;
  __syncthreads();
  if (tid == 0) {
    float tot = 0.0f;
    #pragma unroll
    for (int w = 0; w < 8; ++w) tot += sPart[w];
    atomicAdd(out, -tot);                // loss = kl*N - sum_n lse_n
  }
}

// ---------------- host launch ----------------------------------------------
extern "C" void kernel_launch(void* const* d_in, const int* in_sizes, int n_in,
                              void* d_out, int out_size, void* d_ws, size_t ws_size,
                              hipStream_t stream) {
  const float* X          = (const float*)d_in[0];
  const float* u_noise    = (const float*)d_in[1];
  const float* phi_logits = (const float*)d_in[2];
  const float* q_mu       = (const float*)d_in[3];
  const float* q_logvar   = (const float*)d_in[4];
  const float* pi_logits  = (const float*)d_in[5];
  const float* prior_phi  = (const float*)d_in[6];
  float* out = (float*)d_out;

  float2* wfrag = (float2*)d_ws;                 // 4096 float2 = 32 KB
  float*  wsC   = (float*)d_ws + 8192;           // 64 floats
  float*  wsLp  = wsC + 64;                      // 64 floats

  gmm_setup<<<1, 64, 0, stream>>>(u_noise, phi_logits, q_mu, q_logvar,
                                  pi_logits, prior_phi, out, wfrag, wsC, wsLp);
  gmm_main<<<N_ROWS / 128, 256, 0, stream>>>(X, (const float4*)wfrag,
                                             wsC, wsLp, out);
}